// FM_49701361549558
// MI455X (gfx1250) — compile-verified
//
#include <hip/hip_runtime.h>
#include <math.h>

typedef __attribute__((ext_vector_type(2))) float v2f;
typedef __attribute__((ext_vector_type(8))) float v8f;

#define FM_B        16384
#define FM_F        64
#define ROWS_PER_WAVE   16
#define WAVES_PER_BLOCK 8
#define TPB             256   // 8 wave32

__device__ __forceinline__ float sq4(float4 a) {
    return a.x * a.x + a.y * a.y + a.z * a.z + a.w * a.w;
}

__global__ __launch_bounds__(TPB) void FM_49701361549558_kernel(
    const int*   __restrict__ user_ids,
    const int*   __restrict__ item_ids,
    const int*   __restrict__ metadata_ids,   // [B,2] row-major
    const float* __restrict__ user_emb,
    const float* __restrict__ item_emb,
    const float* __restrict__ meta_emb0,
    const float* __restrict__ meta_emb1,
    const float* __restrict__ user_lin,
    const float* __restrict__ item_lin,
    const float* __restrict__ meta_lin0,
    const float* __restrict__ meta_lin1,
    float*       __restrict__ out)
{
    const int lane = threadIdx.x & 31;
    const int wave = threadIdx.x >> 5;
    const int base = (blockIdx.x * WAVES_PER_BLOCK + wave) * ROWS_PER_WAVE;

    // Lane ownership: row m = lane&15, half h = lane>>4 covers 32 of 64 factors.
    const int m   = lane & 15;
    const int h   = lane >> 4;
    const int row = base + m;

    const int uid  = user_ids[row];
    const int iid  = item_ids[row];
    const int mid0 = metadata_ids[2 * row + 0];
    const int mid1 = metadata_ids[2 * row + 1];

    // Each lane streams 8 x float4 (=32 floats) per table: global_load_b128 gathers.
    const float4* up = reinterpret_cast<const float4*>(user_emb  + (size_t)uid  * FM_F) + h * 8;
    const float4* ip = reinterpret_cast<const float4*>(item_emb  + (size_t)iid  * FM_F) + h * 8;
    const float4* ap = reinterpret_cast<const float4*>(meta_emb0 + (size_t)mid0 * FM_F) + h * 8;
    const float4* bp = reinterpret_cast<const float4*>(meta_emb1 + (size_t)mid1 * FM_F) + h * 8;

    float acc = 0.0f;
#pragma unroll
    for (int c = 0; c < 8; ++c) {
        float4 u = up[c];
        float4 i = ip[c];
        float4 a = ap[c];
        float4 b = bp[c];
        float4 s;
        s.x = u.x + i.x + a.x + b.x;
        s.y = u.y + i.y + a.y + b.y;
        s.z = u.z + i.z + a.z + b.z;
        s.w = u.w + i.w + a.w + b.w;
        acc += 0.5f * (sq4(s) - sq4(u) - sq4(i) - sq4(a) - sq4(b));
    }

    // Exact 2-way per-row reduction via WMMA f32 16x16x4:
    //   A[m][0] = p(lane m), A[m][2] = p(lane 16+m), A[m][1]=A[m][3]=0
    //   B = all ones  =>  D[m][n] = p(lane m) + p(lane 16+m)  (row m's pairwise term)
    // All-ones B makes the result independent of B's exact lane striping.
    v2f A; A.x = acc;  A.y = 0.0f;
    v2f Bm; Bm.x = 1.0f; Bm.y = 1.0f;
    v8f C = {};
    // 8 args: (neg_a, A, neg_b, B, c_mod, C, reuse_a, reuse_b)
    v8f D = __builtin_amdgcn_wmma_f32_16x16x4_f32(
        false, A, false, Bm, (short)0, C, false, false);

    // C/D layout: lanes 0-15 hold rows 0-7 in D[0..7]; lanes 16-31 hold rows 8-15.
    const int idx = lane & 7;
    float pair = D[0];
#pragma unroll
    for (int k = 1; k < 8; ++k)
        if (idx == k) pair = D[k];

    const bool w1 = (lane < 8);                  // writes rows base+0..7
    const bool w2 = (lane >= 16) && (lane < 24); // writes rows base+8..15
    if (w1 || w2) {
        const int r    = w1 ? lane : (lane - 8);
        const int orow = base + r;
        const int u2 = user_ids[orow];
        const int i2 = item_ids[orow];
        const int a2 = metadata_ids[2 * orow + 0];
        const int b2 = metadata_ids[2 * orow + 1];
        const float lin = user_lin[u2] + item_lin[i2] + meta_lin0[a2] + meta_lin1[b2];
        const float x = lin + pair;
        out[orow] = 1.0f / (1.0f + expf(-x));
    }
}

extern "C" void kernel_launch(void* const* d_in, const int* in_sizes, int n_in,
                              void* d_out, int out_size, void* d_ws, size_t ws_size,
                              hipStream_t stream) {
    (void)in_sizes; (void)n_in; (void)d_ws; (void)ws_size; (void)out_size;

    const int*   user_ids     = (const int*)  d_in[0];
    const int*   item_ids     = (const int*)  d_in[1];
    const int*   metadata_ids = (const int*)  d_in[2];
    const float* user_emb     = (const float*)d_in[3];
    const float* item_emb     = (const float*)d_in[4];
    const float* meta_emb0    = (const float*)d_in[5];
    const float* meta_emb1    = (const float*)d_in[6];
    const float* user_lin     = (const float*)d_in[7];
    const float* item_lin     = (const float*)d_in[8];
    const float* meta_lin0    = (const float*)d_in[9];
    const float* meta_lin1    = (const float*)d_in[10];
    float*       out          = (float*)d_out;

    // 16384 rows / (16 rows per wave * 8 waves per block) = 128 blocks, exact.
    const int grid = FM_B / (ROWS_PER_WAVE * WAVES_PER_BLOCK);
    FM_49701361549558_kernel<<<grid, TPB, 0, stream>>>(
        user_ids, item_ids, metadata_ids,
        user_emb, item_emb, meta_emb0, meta_emb1,
        user_lin, item_lin, meta_lin0, meta_lin1, out);
}